// PhotoreceptorLayer_91156385890749
// MI455X (gfx1250) — compile-verified
//
#include <hip/hip_runtime.h>
#include <math.h>

// Problem shape (fixed by the reference): inputs (B=16, U=2, T=2000, D=1024) f32.
#define T_STEPS 2000
#define D_DIM   1024
#define DT      0.008f          // TIME_STEP = 0.001 * TIME_BIN, TIME_BIN = 8
#define DEPTH   16              // async-to-LDS prefetch depth (power of two)
#define BLOCK   64              // 2 wave32s per workgroup -> maximal WGP spread

// ---------------------------------------------------------------------------
// Templated scan body. FASTPOW=true when cgmphill==3 && hillcoef==4 (uniform
// per unit): powf -> multiplies, IEEE div -> v_rcp_f32.
// Input staging: per-wave LDS ring (DEPTH slots x 32 lanes x 4B) filled with
// gfx1250 async-to-LDS copies (ASYNCcnt), consumed with ds_load_b32 (DScnt).
// Pipeline per iteration t: [wait asynccnt<=DEPTH-1; ds_load x[t]] -> compute
// with x[t-1] -> [wait dscnt 0; re-arm slot with x[t+DEPTH]].  The ds_load
// latency hides under the compute; in-order ASYNC completion makes the
// constant wait sufficient.  Tail iterations drain with s_wait_asynccnt 0.
// ---------------------------------------------------------------------------
template <bool FASTPOW>
__device__ __forceinline__ void rieke_run(
    const float* __restrict__ gx,     // &x[bu*T*D + d]  (stride D per step)
    float* __restrict__ gout,         // &out[bu*T*D + d]
    float sigma, float phi, float eta, float beta, float gamma,
    float cgmphill, float hillcoef,
    float gdark_eff, float cdark, float inv_ha,
    float kA, float out_scale, float smax,
    float k_r, float k_p, float k_e, float k_c,
    float* ring, unsigned lds_lane_base, int ring_base)
{
    // ---- prologue: async-stage x[0..DEPTH-1] into the LDS ring ----
    {
        unsigned long long gp = (unsigned long long)(const void*)gx;
        #pragma unroll
        for (int k = 0; k < DEPTH; ++k) {
            unsigned la = lds_lane_base + (unsigned)(k * 128);
            asm volatile("global_load_async_to_lds_b32 %0, %1, off"
                         :: "v"(la), "v"(gp) : "memory");
            gp += (unsigned long long)D_DIM * 4ull;
        }
    }

    // consume x[0] (in-order completion: <=DEPTH-1 outstanding => x[0] landed)
    asm volatile("s_wait_asynccnt %0" :: "n"(DEPTH - 1) : "memory");
    float x_prev = ring[ring_base + 0];
    asm volatile("s_wait_dscnt 0" ::: "memory");

    // re-arm slot 0 with x[DEPTH]
    unsigned long long gnext =
        (unsigned long long)(const void*)(gx) + (unsigned long long)DEPTH * D_DIM * 4ull;
    asm volatile("global_load_async_to_lds_b32 %0, %1, off"
                 :: "v"(lds_lane_base), "v"(gnext) : "memory");
    gnext += (unsigned long long)D_DIM * 4ull;

    // ---- initial state (matches reference) ----
    float r = x_prev * gamma / sigma;
    float p = (eta + r) / phi;
    float g = gdark_eff;
    float s = gdark_eff * eta / phi;
    float c = cdark;
    float gp_pow = FASTPOW ? (g * g * g) : powf(g, cgmphill);

    // out[t=0]: reference concatenates g=0 at t=0
    {
        float z = FASTPOW ? 0.0f : powf(0.0f, cgmphill);
        __builtin_nontemporal_store(out_scale * z, gout);
    }

    // ---- steady-state: t = 1 .. T-1-DEPTH (all waits/re-arms unconditional) --
    #pragma unroll 16
    for (int t = 1; t <= T_STEPS - 1 - DEPTH; ++t) {
        const int slot = t & (DEPTH - 1);

        // start fetching x[t] (needed next iteration); latency hides below
        asm volatile("s_wait_asynccnt %0" :: "n"(DEPTH - 1) : "memory");
        float x_next = ring[ring_base + slot * 32];

        // step t: consumes x[t-1]   (FMA-folded Euler updates)
        float r_n = k_r * r + gamma * x_prev;          // r + dt(-sigma r) + g x
        float p_n = k_p * p + (DT * r + k_e);          // p + dt(r + eta - phi p)
        float c_n = k_c * c + kA * gp_pow;             // c + dt(cur2ca*I/2 - b c)
        float hc  = c_n * inv_ha;
        float hp;
        if (FASTPOW) { float h2 = hc * hc; hp = h2 * h2; }
        else         { hp = powf(hc, hillcoef); }
        float s_n;
        if (FASTPOW) s_n = smax * __builtin_amdgcn_rcpf(1.0f + hp);
        else         s_n = smax / (1.0f + hp);
        float g_n = g + DT * (s - p * g);
        float gpn = FASTPOW ? (g_n * g_n * g_n) : powf(g_n, cgmphill);

        __builtin_nontemporal_store(out_scale * gpn, gout + (size_t)t * D_DIM);

        r = r_n; p = p_n; c = c_n; s = s_n; g = g_n; gp_pow = gpn;

        // x_next has landed by now; re-arm this slot with x[t+DEPTH]
        asm volatile("s_wait_dscnt 0" ::: "memory");
        unsigned la = lds_lane_base + (unsigned)(slot * 128);
        asm volatile("global_load_async_to_lds_b32 %0, %1, off"
                     :: "v"(la), "v"(gnext) : "memory");
        gnext += (unsigned long long)D_DIM * 4ull;
        x_prev = x_next;
    }

    // ---- tail: t = T-DEPTH .. T-1 (drain ring; wait for ALL outstanding) ----
    #pragma unroll
    for (int t = T_STEPS - DEPTH; t < T_STEPS; ++t) {
        float r_n = k_r * r + gamma * x_prev;
        float p_n = k_p * p + (DT * r + k_e);
        float c_n = k_c * c + kA * gp_pow;
        float hc  = c_n * inv_ha;
        float hp;
        if (FASTPOW) { float h2 = hc * hc; hp = h2 * h2; }
        else         { hp = powf(hc, hillcoef); }
        float s_n;
        if (FASTPOW) s_n = smax * __builtin_amdgcn_rcpf(1.0f + hp);
        else         s_n = smax / (1.0f + hp);
        float g_n = g + DT * (s - p * g);
        float gpn = FASTPOW ? (g_n * g_n * g_n) : powf(g_n, cgmphill);

        __builtin_nontemporal_store(out_scale * gpn, gout + (size_t)t * D_DIM);

        r = r_n; p = p_n; c = c_n; s = s_n; g = g_n; gp_pow = gpn;

        if (t < T_STEPS - 1) {
            asm volatile("s_wait_asynccnt 0" ::: "memory");  // everything landed
            x_prev = ring[ring_base + (t & (DEPTH - 1)) * 32];
        }
    }
}

__global__ __launch_bounds__(BLOCK) void PhotoreceptorLayer_91156385890749_kernel(
    const float* __restrict__ x,
    const float* __restrict__ sigma_p,    const float* __restrict__ phi_p,
    const float* __restrict__ eta_p,      const float* __restrict__ beta_p,
    const float* __restrict__ gamma_p,    const float* __restrict__ cgmp2cur_p,
    const float* __restrict__ cgmphill_p, const float* __restrict__ cdark_p,
    const float* __restrict__ hillcoef_p, const float* __restrict__ hillaff_p,
    const float* __restrict__ gdark_p,
    float* __restrict__ out)
{
    __shared__ float ring[(BLOCK / 32) * DEPTH * 32];

    // grid = (D/BLOCK, B, U): no integer division needed for the unit index
    const int u    = blockIdx.z;
    const int bu   = blockIdx.y * gridDim.z + blockIdx.z;   // b-major (B,U)
    const int lane = threadIdx.x & 31;
    const int wave = threadIdx.x >> 5;
    const int d    = blockIdx.x * BLOCK + threadIdx.x;

    // per-unit scaled parameters (uniform per block -> SGPRs)
    const float sigma    = sigma_p[u] * 10.0f;
    const float phi      = phi_p[u] * 10.0f;
    const float eta      = eta_p[u] * 1000.0f;
    const float beta     = beta_p[u] * 10.0f;
    const float gamma    = gamma_p[u] * 10.0f / 8.0f;   // gamma*10/TIME_BIN
    const float cgmp2cur = cgmp2cur_p[u];
    const float cgmphill = cgmphill_p[u];
    const float cdark    = cdark_p[u];
    const float hillcoef = hillcoef_p[u];
    const float hillaff  = hillaff_p[u];
    const float gdark    = gdark_p[u] * 100.0f;

    // derived constants (once per block)
    const float darkCurrent = powf(gdark, cgmphill) * cgmp2cur * 0.5f;
    const float gdark_eff   = powf(2.0f * darkCurrent / cgmp2cur, 1.0f / cgmphill);
    const float cur2ca      = beta * cdark / darkCurrent;
    const float smax        = eta / phi * gdark_eff *
                              (1.0f + powf(cdark / hillaff, hillcoef));
    const float inv_ha      = 1.0f / hillaff;
    const float out_scale   = -cgmp2cur * 0.5f;          // out = -(cgmp2cur*g^h)/2

    // FMA-folded Euler coefficients (uniform -> SGPRs)
    const float kA  = DT * (cur2ca * cgmp2cur * 0.5f);
    const float k_r = 1.0f - DT * sigma;
    const float k_p = 1.0f - DT * phi;
    const float k_e = DT * eta;
    const float k_c = 1.0f - DT * beta;

    const size_t base = (size_t)bu * (size_t)T_STEPS * D_DIM + (size_t)d;
    const float* gx   = x + base;
    float*       gout = out + base;

    // LDS byte offset of this lane's slot-0 entry in its wave's ring region
    const unsigned lds_lane_base =
        (unsigned)(size_t)(&ring[0]) + (unsigned)((wave * DEPTH * 32 + lane) * 4);
    const int ring_base = wave * DEPTH * 32 + lane;

    const bool fast = (cgmphill == 3.0f) && (hillcoef == 4.0f);
    if (fast)
        rieke_run<true >(gx, gout, sigma, phi, eta, beta, gamma, cgmphill, hillcoef,
                         gdark_eff, cdark, inv_ha, kA, out_scale, smax,
                         k_r, k_p, k_e, k_c, ring, lds_lane_base, ring_base);
    else
        rieke_run<false>(gx, gout, sigma, phi, eta, beta, gamma, cgmphill, hillcoef,
                         gdark_eff, cdark, inv_ha, kA, out_scale, smax,
                         k_r, k_p, k_e, k_c, ring, lds_lane_base, ring_base);
}

extern "C" void kernel_launch(void* const* d_in, const int* in_sizes, int n_in,
                              void* d_out, int out_size, void* d_ws, size_t ws_size,
                              hipStream_t stream) {
    (void)n_in; (void)out_size; (void)d_ws; (void)ws_size;
    const float* x        = (const float*)d_in[0];
    const float* sigma    = (const float*)d_in[1];
    const float* phi      = (const float*)d_in[2];
    const float* eta      = (const float*)d_in[3];
    const float* beta     = (const float*)d_in[4];
    const float* gamma    = (const float*)d_in[5];
    const float* cgmp2cur = (const float*)d_in[6];
    const float* cgmphill = (const float*)d_in[7];
    const float* cdark    = (const float*)d_in[8];
    const float* hillcoef = (const float*)d_in[9];
    const float* hillaff  = (const float*)d_in[10];
    const float* gdark    = (const float*)d_in[11];
    float* out = (float*)d_out;

    const int U  = in_sizes[1];                               // 2
    const int BU = in_sizes[0] / (T_STEPS * D_DIM);           // B*U = 32
    const int B  = BU / U;                                    // 16

    dim3 grid(D_DIM / BLOCK, B, U);                           // (16, 16, 2)
    PhotoreceptorLayer_91156385890749_kernel<<<grid, BLOCK, 0, stream>>>(
        x, sigma, phi, eta, beta, gamma, cgmp2cur, cgmphill, cdark,
        hillcoef, hillaff, gdark, out);
}